// GCN_91311004713379
// MI455X (gfx1250) — compile-verified
//
#include <hip/hip_runtime.h>

typedef __attribute__((ext_vector_type(2))) float v2f;
typedef __attribute__((ext_vector_type(8))) float v8f;

#define NN   50000
#define EE   800000
#define GGR  64
#define FIN  512
#define FH   64
#define FOUT 10
#define SCAN_B 512

// ---------------------------------------------------------------- misc utils
__global__ void k_zero_int(int* __restrict__ p, int n) {
    int i = blockIdx.x * blockDim.x + threadIdx.x;
    if (i < n) p[i] = 0;
}

// ---------------------------------------------------------------- degree/norm
__global__ void k_deg_count(const int* __restrict__ dst, int* __restrict__ edeg, int e) {
    int i = blockIdx.x * blockDim.x + threadIdx.x;
    if (i < e) atomicAdd(&edeg[dst[i]], 1);
}

__global__ void k_dinv(const int* __restrict__ edeg, float* __restrict__ dinv, int n) {
    int i = blockIdx.x * blockDim.x + threadIdx.x;
    if (i < n) dinv[i] = rsqrtf((float)(edeg[i] + 1));  // +1 self-loop
}

__global__ void k_norm(const int* __restrict__ src, const int* __restrict__ dst,
                       const float* __restrict__ dinv, float* __restrict__ norm, int e) {
    int i = blockIdx.x * blockDim.x + threadIdx.x;
    if (i < e) norm[i] = dinv[src[i]] * dinv[dst[i]];
}

// ---------------------------------------------------------------- CSR build
// Exclusive scan of edeg -> rowptr.  Level 1: per-block LDS scan (512 wide).
__global__ __launch_bounds__(SCAN_B) void k_scan_block(const int* __restrict__ in,
                                                       int* __restrict__ ex,
                                                       int* __restrict__ bsums, int n) {
    __shared__ int sh[SCAN_B];
    const int t = threadIdx.x;
    const int i = blockIdx.x * SCAN_B + t;
    int v = (i < n) ? in[i] : 0;
    sh[t] = v;
    __syncthreads();
    for (int off = 1; off < SCAN_B; off <<= 1) {
        int x = (t >= off) ? sh[t - off] : 0;
        __syncthreads();
        sh[t] += x;
        __syncthreads();
    }
    if (i < n) ex[i] = sh[t] - v;                 // exclusive within block
    if (t == SCAN_B - 1) bsums[blockIdx.x] = sh[t];
}

// Level 2: serial exclusive scan of ~98 block sums (negligible).
__global__ void k_scan_tops(int* __restrict__ bsums, int nb) {
    if (blockIdx.x == 0 && threadIdx.x == 0) {
        int run = 0;
        for (int i = 0; i < nb; ++i) { int v = bsums[i]; bsums[i] = run; run += v; }
    }
}

__global__ void k_scan_add(int* __restrict__ ex, const int* __restrict__ bsums, int n) {
    int i = blockIdx.x * blockDim.x + threadIdx.x;
    if (i < n) ex[i] += bsums[i / SCAN_B];
}

__global__ void k_csr_fill(const int* __restrict__ src, const int* __restrict__ dst,
                           const float* __restrict__ nrm, const int* __restrict__ rowptr,
                           int* __restrict__ cursor, int* __restrict__ csr_src,
                           float* __restrict__ csr_nrm, int e) {
    int i = blockIdx.x * blockDim.x + threadIdx.x;
    if (i < e) {
        int d = dst[i];
        int slot = atomicAdd(&cursor[d], 1);
        int p = rowptr[d] + slot;
        csr_src[p] = src[i];
        csr_nrm[p] = nrm[i];
    }
}

// ---------------------------------------------------------------- W packing
// Wp[idx], idx = s*256 + lane*8 + t*2 + j  ->  W[(4s + 2*(lane>>4) + j)*64 + t*16 + (lane&15)]
// so each lane fetches its four B fragments per K-step with two b128 loads.
__global__ void k_packW(const float* __restrict__ W, float* __restrict__ Wp, int K) {
    int idx = blockIdx.x * blockDim.x + threadIdx.x;
    if (idx < K * 64) {
        int j    = idx & 1;
        int t    = (idx >> 1) & 3;
        int lane = (idx >> 3) & 31;
        int s    = idx >> 8;
        int k    = 4 * s + 2 * (lane >> 4) + j;
        int n    = t * 16 + (lane & 15);
        Wp[idx] = W[k * FH + n];
    }
}

// ---------------------------------------------------------------- WMMA GEMM
// H[M x 64] = X[M x K] @ W[K x 64] via V_WMMA_F32_16X16X4_F32 (fp32-exact).
// One wave -> 16 rows x 64 cols; 256-thread block = 8 waves = 128 rows.
template <int K>
__global__ __launch_bounds__(256) void k_gemm(const float* __restrict__ X,
                                              const float* __restrict__ Wp,
                                              float* __restrict__ H, int M) {
    const int wave = threadIdx.x >> 5;
    const int lane = threadIdx.x & 31;
    const int half = lane >> 4;
    const int mrow = lane & 15;

    const int tile_m = (blockIdx.x * 8 + wave) * 16;
    int m = tile_m + mrow;
    int mload = (m < M) ? m : (M - 1);              // clamp: keep EXEC all-ones for WMMA
    const float* __restrict__ xrow = X + (size_t)mload * K;
    const float* __restrict__ wp   = Wp + lane * 8;

    v8f acc0 = {}, acc1 = {}, acc2 = {}, acc3 = {};

    for (int s = 0; s < K / 4; ++s) {
        const float2 av = *(const float2*)(xrow + 4 * s + 2 * half);
        v2f a = {av.x, av.y};

        const float4 p0 = *(const float4*)(wp + s * 256);
        const float4 p1 = *(const float4*)(wp + s * 256 + 4);
        v2f b0 = {p0.x, p0.y}, b1 = {p0.z, p0.w};
        v2f b2 = {p1.x, p1.y}, b3 = {p1.z, p1.w};

        acc0 = __builtin_amdgcn_wmma_f32_16x16x4_f32(false, a, false, b0, (short)0, acc0, false, false);
        acc1 = __builtin_amdgcn_wmma_f32_16x16x4_f32(false, a, false, b1, (short)0, acc1, false, false);
        acc2 = __builtin_amdgcn_wmma_f32_16x16x4_f32(false, a, false, b2, (short)0, acc2, false, false);
        acc3 = __builtin_amdgcn_wmma_f32_16x16x4_f32(false, a, false, b3, (short)0, acc3, false, false);
    }

    // D layout: VGPR r, lane half h, col n -> D[M = r + 8h][N = n]
#pragma unroll
    for (int r = 0; r < 8; ++r) {
        int row = tile_m + r + 8 * half;
        if (row < M) {
            float* __restrict__ orow = H + (size_t)row * FH + mrow;
            orow[0]  = acc0[r];
            orow[16] = acc1[r];
            orow[32] = acc2[r];
            orow[48] = acc3[r];
        }
    }
}

// ---------------------------------------------------------------- aggregation
// Atomic-free gather: 16 lanes per node (4 feats each) walk the node's in-edges.
// OUT[v][f] = b[f] + dinv[v]^2 * H[v][f] + sum_e norm[e] * H[src[e]][f]
__global__ void k_agg_csr(const int* __restrict__ rowptr, const int* __restrict__ edeg,
                          const int* __restrict__ csr_src, const float* __restrict__ csr_nrm,
                          const float* __restrict__ H, const float* __restrict__ dinv,
                          const float* __restrict__ b, float* __restrict__ OUT, int n) {
    int t = blockIdx.x * blockDim.x + threadIdx.x;
    int node = t >> 4;
    int q = (t & 15) * 4;
    if (node >= n) return;

    float dv = dinv[node];
    float w  = dv * dv;
    const float4 hb = *(const float4*)(b + q);
    const float4 hs = *(const float4*)(H + (size_t)node * FH + q);
    float s0 = hb.x + hs.x * w;
    float s1 = hb.y + hs.y * w;
    float s2 = hb.z + hs.z * w;
    float s3 = hb.w + hs.w * w;

    const int beg = rowptr[node];
    const int cnt = edeg[node];
    for (int j = 0; j < cnt; ++j) {
        int srcn = csr_src[beg + j];
        float nv = csr_nrm[beg + j];
        const float4 hv = *(const float4*)(H + (size_t)srcn * FH + q);
        s0 += hv.x * nv;
        s1 += hv.y * nv;
        s2 += hv.z * nv;
        s3 += hv.w * nv;
    }
    float4 o = {s0, s1, s2, s3};
    *(float4*)(OUT + (size_t)node * FH + q) = o;
}

// -------------------------------------------------- pooling + head
__global__ void k_pool_clear(float* __restrict__ sums, float* __restrict__ cnts) {
    int i = blockIdx.x * blockDim.x + threadIdx.x;
    if (i < GGR * FH) sums[i] = 0.0f;
    if (i < GGR) cnts[i] = 0.0f;
}

__global__ void k_pool(const float* __restrict__ H, const int* __restrict__ batch,
                       float* __restrict__ sums, float* __restrict__ cnts, int n) {
    int t = blockIdx.x * blockDim.x + threadIdx.x;
    int node = t >> 6, f = t & 63;
    if (node < n) {
        int g = batch[node];
        atomicAdd(&sums[g * FH + f], H[t]);
        if (f == 0) atomicAdd(&cnts[g], 1.0f);
    }
}

__global__ void k_head(const float* __restrict__ sums, const float* __restrict__ cnts,
                       const float* __restrict__ Wlin, const float* __restrict__ blin,
                       float* __restrict__ out) {
    int t = blockIdx.x * blockDim.x + threadIdx.x;
    if (t < GGR * FOUT) {
        int g = t / FOUT, o = t % FOUT;
        float c = fmaxf(cnts[g], 1.0f);
        float acc = 0.0f;
        for (int f = 0; f < FH; ++f) acc += sums[g * FH + f] * Wlin[f * FOUT + o];
        out[t] = acc / c + blin[o];
    }
}

// ---------------------------------------------------------------- launch
extern "C" void kernel_launch(void* const* d_in, const int* in_sizes, int n_in,
                              void* d_out, int out_size, void* d_ws, size_t ws_size,
                              hipStream_t stream) {
    const float* x    = (const float*)d_in[0];
    const int*   ei   = (const int*)d_in[1];   // [2, E] int32 (JAX x64 off)
    const int*   bat  = (const int*)d_in[2];
    const float* W1   = (const float*)d_in[3];
    const float* b1   = (const float*)d_in[4];
    const float* W2   = (const float*)d_in[5];
    const float* b2   = (const float*)d_in[6];
    const float* W3   = (const float*)d_in[7];
    const float* b3   = (const float*)d_in[8];
    const float* Wlin = (const float*)d_in[9];
    const float* blin = (const float*)d_in[10];
    float* out = (float*)d_out;

    const int* src = ei;
    const int* dst = ei + EE;

    // workspace carve-up (bytes), 256-aligned
    char* ws = (char*)d_ws;
    size_t off = 0;
    auto carve = [&](size_t bytes) { void* p = ws + off; off = (off + bytes + 255) & ~(size_t)255; return p; };
    int*   edeg    = (int*)  carve(sizeof(int)   * NN);
    float* dinv    = (float*)carve(sizeof(float) * NN);
    float* nrm     = (float*)carve(sizeof(float) * EE);
    int*   rowptr  = (int*)  carve(sizeof(int)   * NN);
    int*   bsums   = (int*)  carve(sizeof(int)   * 256);
    int*   cursor  = (int*)  carve(sizeof(int)   * NN);
    int*   csr_src = (int*)  carve(sizeof(int)   * EE);
    float* csr_nrm = (float*)carve(sizeof(float) * EE);
    float* bufA    = (float*)carve(sizeof(float) * (size_t)NN * FH);  // GEMM out
    float* bufB    = (float*)carve(sizeof(float) * (size_t)NN * FH);  // aggregated
    float* Wp1     = (float*)carve(sizeof(float) * FIN * FH);
    float* Wp2     = (float*)carve(sizeof(float) * FH * FH);
    float* Wp3     = (float*)carve(sizeof(float) * FH * FH);
    float* sums    = (float*)carve(sizeof(float) * GGR * FH);
    float* cnts    = (float*)carve(sizeof(float) * GGR);
    (void)ws_size; (void)in_sizes; (void)n_in; (void)out_size;

    const int TB = 256;
    const int nScanBlk = (NN + SCAN_B - 1) / SCAN_B;      // 98
    const int gemmGrid = (NN + 127) / 128;

    // graph normalization
    k_zero_int <<<(NN + TB - 1) / TB, TB, 0, stream>>>(edeg, NN);
    k_deg_count<<<(EE + TB - 1) / TB, TB, 0, stream>>>(dst, edeg, EE);
    k_dinv     <<<(NN + TB - 1) / TB, TB, 0, stream>>>(edeg, dinv, NN);
    k_norm     <<<(EE + TB - 1) / TB, TB, 0, stream>>>(src, dst, dinv, nrm, EE);

    // CSR build (once, reused for all 3 layers)
    k_scan_block<<<nScanBlk, SCAN_B, 0, stream>>>(edeg, rowptr, bsums, NN);
    k_scan_tops <<<1, 1, 0, stream>>>(bsums, nScanBlk);
    k_scan_add  <<<(NN + TB - 1) / TB, TB, 0, stream>>>(rowptr, bsums, NN);
    k_zero_int  <<<(NN + TB - 1) / TB, TB, 0, stream>>>(cursor, NN);
    k_csr_fill  <<<(EE + TB - 1) / TB, TB, 0, stream>>>(src, dst, nrm, rowptr, cursor,
                                                        csr_src, csr_nrm, EE);

    // pack weights into WMMA-fragment order
    k_packW<<<(FIN * FH + TB - 1) / TB, TB, 0, stream>>>(W1, Wp1, FIN);
    k_packW<<<(FH  * FH + TB - 1) / TB, TB, 0, stream>>>(W2, Wp2, FH);
    k_packW<<<(FH  * FH + TB - 1) / TB, TB, 0, stream>>>(W3, Wp3, FH);

    const int aggThreads = NN * 16;

    // layer 1
    k_gemm<FIN><<<gemmGrid, TB, 0, stream>>>(x, Wp1, bufA, NN);
    k_agg_csr  <<<(aggThreads + TB - 1) / TB, TB, 0, stream>>>(rowptr, edeg, csr_src, csr_nrm,
                                                               bufA, dinv, b1, bufB, NN);
    // layer 2
    k_gemm<FH><<<gemmGrid, TB, 0, stream>>>(bufB, Wp2, bufA, NN);
    k_agg_csr <<<(aggThreads + TB - 1) / TB, TB, 0, stream>>>(rowptr, edeg, csr_src, csr_nrm,
                                                              bufA, dinv, b2, bufB, NN);
    // layer 3
    k_gemm<FH><<<gemmGrid, TB, 0, stream>>>(bufB, Wp3, bufA, NN);
    k_agg_csr <<<(aggThreads + TB - 1) / TB, TB, 0, stream>>>(rowptr, edeg, csr_src, csr_nrm,
                                                              bufA, dinv, b3, bufB, NN);

    // pool + head
    k_pool_clear<<<(GGR * FH + TB - 1) / TB, TB, 0, stream>>>(sums, cnts);
    k_pool      <<<((NN * FH) + TB - 1) / TB, TB, 0, stream>>>(bufB, bat, sums, cnts, NN);
    k_head      <<<(GGR * FOUT + TB - 1) / TB, TB, 0, stream>>>(sums, cnts, Wlin, blin, out);
}